// ResGAT_3856880632465
// MI455X (gfx1250) — compile-verified
//
#include <hip/hip_runtime.h>

#define NN 23   // graph nodes
#define KP 32   // padded k extent (two 16-wide WMMA N-tiles)
#define MP 24   // padded m extent (six 4-wide WMMA K-steps)
#define TT 12   // time steps; we only use slice TT-1

typedef float v2f __attribute__((ext_vector_type(2)));
typedef float v8f __attribute__((ext_vector_type(8)));

// ---------------------------------------------------------------------------
// Prep kernel: build zero-padded folded attention matrices + residual weight.
//   A1p[i, k, m] = att1[i,k,m] * yzin[i,k,0] * yzout[i,k,1]   (k<23, m<23)
//   A2p[i, k, m] = att2[i,k,m] * yzin[i,k,1] * yzout[i,k,1]
//   zeros for k in [23,32) or m == 23   -> no masking in the hot loop
//   c0[i,k] = yzout[i,k,0]
// One thread per (i, padded-k) pair: NN*KP = 736 threads.
// ---------------------------------------------------------------------------
__global__ void resgat_prep(const int* __restrict__ adj,
                            const float* __restrict__ W1,
                            const float* __restrict__ W2,
                            const float* __restrict__ a_inner,
                            const float* __restrict__ a_outer,
                            float* __restrict__ A1p,
                            float* __restrict__ A2p,
                            float* __restrict__ c0)
{
    int t = blockIdx.x * blockDim.x + threadIdx.x;
    if (t >= NN * KP) return;
    int i = t / KP;
    int k = t % KP;

    float* o1 = A1p + (long long)t * MP;
    float* o2 = A2p + (long long)t * MP;

    if (k >= NN) {              // padded k rows: all zero
#pragma unroll
        for (int m = 0; m < MP; ++m) { o1[m] = 0.0f; o2[m] = 0.0f; }
        return;
    }

    int idx = i * NN + k;       // index into the unpadded [N,N] tables

    float ai0 = a_inner[idx * 2 + 0];
    float ai1 = a_inner[idx * 2 + 1];
    float yi0 = ai0 / (ai0 + ai1);
    float yi1 = ai1 / (ai0 + ai1);

    float ao0 = a_outer[idx * 2 + 0];
    float ao1 = a_outer[idx * 2 + 1];
    float yo0 = ao0 / (ao0 + ao1);
    float yo1 = ao1 / (ao0 + ao1);

    c0[idx] = yo0;

    float msk[NN];
#pragma unroll
    for (int m = 0; m < NN; ++m)
        msk[m] = (adj[k * NN + m] != 0 || m == k) ? 1.0f : 0.0f;   // diag forced 1

    const float* w1 = W1 + (long long)idx * NN;
    const float* w2 = W2 + (long long)idx * NN;

    // branch 1 (att over W1 row)
    {
        float mn = w1[0];
#pragma unroll
        for (int m = 1; m < NN; ++m) mn = fminf(mn, w1[m]);
        mn = (mn > 0.0f) ? 0.0f : mn;
        float att[NN]; float s = 0.0f;
#pragma unroll
        for (int m = 0; m < NN; ++m) { att[m] = (w1[m] - mn) * msk[m]; s += att[m]; }
        float scale = yi0 * yo1 / s;
#pragma unroll
        for (int m = 0; m < NN; ++m) o1[m] = att[m] * scale;
        o1[NN] = 0.0f;          // m = 23 pad
    }
    // branch 2 (att over W2 row)
    {
        float mn = w2[0];
#pragma unroll
        for (int m = 1; m < NN; ++m) mn = fminf(mn, w2[m]);
        mn = (mn > 0.0f) ? 0.0f : mn;
        float att[NN]; float s = 0.0f;
#pragma unroll
        for (int m = 0; m < NN; ++m) { att[m] = (w2[m] - mn) * msk[m]; s += att[m]; }
        float scale = yi1 * yo1 / s;
#pragma unroll
        for (int m = 0; m < NN; ++m) o2[m] = att[m] * scale;
        o2[NN] = 0.0f;          // m = 23 pad
    }
}

// ---------------------------------------------------------------------------
// Main kernel: one wave32 per (16-batch tile, i).
//   out[b,i,k] = d[b,i,k]*c0[i,k] + Sum_m A1p[i,k,m]*d[b,m,i]
//                                 + Sum_m A2p[i,k,m]*d[b,i,m]
// Chained V_WMMA_F32_16X16X4_F32: M=16 batches, N=2x16 k-tiles, K=6x4 m.
// B operands come from the zero-padded workspace as single b64 loads.
// Only the final K-step's upper-lane element (m=23) needs masking on A.
// ---------------------------------------------------------------------------
__global__ __launch_bounds__(256)
void resgat_main(const float* __restrict__ flow,
                 const float* __restrict__ A1p,
                 const float* __restrict__ A2p,
                 const float* __restrict__ c0,
                 float* __restrict__ out,
                 int B, int totalTiles)
{
    int wave = (int)((blockIdx.x * blockDim.x + threadIdx.x) >> 5);
    if (wave >= totalTiles) return;           // wave-uniform: EXEC stays all-ones
    int lane = threadIdx.x & 31;

    int i  = wave % NN;
    int b0 = (wave / NN) * 16;

    int Ml  = lane & 15;      // A: row M ; B: col N ; C: col N
    int hi  = lane >> 4;      // lane half selects K pair {0,1} vs {2,3}
    int klo = hi * 2;

    int bM = b0 + Ml;
    if (bM > B - 1) bM = B - 1;               // clamp (stores guarded later)

    // Per-lane A-side base pointers with klo folded in; per-step offsets are
    // then compile-time immediates (step stride: 4*NN*TT and 4*TT elements).
    //   x1: d[bM, m, i]  -> p1[m * NN*TT]
    //   x2: d[bM, i, m]  -> p2[m * TT]
    const float* p1 = flow + (((long long)bM * NN + klo) * NN + i) * TT + (TT - 1);
    const float* p2 = flow + (((long long)bM * NN + i) * NN + klo) * TT + (TT - 1);

    // B-side: rows are 24 floats (96 B); m0 = 4s+klo is even -> 8B aligned b64.
    const float* B1lo = A1p + ((long long)i * KP + Ml)        * MP + klo;
    const float* B1hi = A1p + ((long long)i * KP + 16 + Ml)   * MP + klo;
    const float* B2lo = A2p + ((long long)i * KP + Ml)        * MP + klo;
    const float* B2hi = A2p + ((long long)i * KP + 16 + Ml)   * MP + klo;

    v8f acc0 = {};   // k in [0,16)
    v8f acc1 = {};   // k in [16,32), upper part zero-padded

    // Steps 0..4: all indices in range, no masking anywhere.
#pragma unroll
    for (int s = 0; s < 5; ++s) {
        const int m = s * 4;                   // + klo per lane via base ptrs
        v2f a1, a2;
        a1.x = p1[(m + 0) * (NN * TT)];
        a1.y = p1[(m + 1) * (NN * TT)];
        a2.x = p2[(m + 0) * TT];
        a2.y = p2[(m + 1) * TT];
        v2f b10 = *(const v2f*)(B1lo + m);
        v2f b11 = *(const v2f*)(B1hi + m);
        v2f b20 = *(const v2f*)(B2lo + m);
        v2f b21 = *(const v2f*)(B2hi + m);

        acc0 = __builtin_amdgcn_wmma_f32_16x16x4_f32(false, a1, false, b10, (short)0, acc0, false, false);
        acc0 = __builtin_amdgcn_wmma_f32_16x16x4_f32(false, a2, false, b20, (short)0, acc0, false, false);
        acc1 = __builtin_amdgcn_wmma_f32_16x16x4_f32(false, a1, false, b11, (short)0, acc1, false, false);
        acc1 = __builtin_amdgcn_wmma_f32_16x16x4_f32(false, a2, false, b21, (short)0, acc1, false, false);
    }

    // Step 5 (m = 20+klo, 21+klo): upper lanes hit m=23 on the .y element ->
    // address not clamped, zero via select. B side needs nothing (padded zeros).
    {
        const int m = 20;
        float f1   = (klo == 0) ? 1.0f : 0.0f;          // m1 = 21+klo < 23 ?
        int   m1el = (klo == 0) ? (m + 1) : (m);        // offset 21 after the add below
        v2f a1, a2;
        a1.x = p1[m * (NN * TT)];
        a1.y = p1[m1el * (NN * TT) + (klo ? NN * TT : 0)] * f1;  // m=21 or 23 (x0)
        a2.x = p2[m * TT];
        a2.y = p2[m1el * TT + (klo ? TT : 0)] * f1;
        v2f b10 = *(const v2f*)(B1lo + m);
        v2f b11 = *(const v2f*)(B1hi + m);
        v2f b20 = *(const v2f*)(B2lo + m);
        v2f b21 = *(const v2f*)(B2hi + m);

        acc0 = __builtin_amdgcn_wmma_f32_16x16x4_f32(false, a1, false, b10, (short)0, acc0, false, false);
        acc0 = __builtin_amdgcn_wmma_f32_16x16x4_f32(false, a2, false, b20, (short)0, acc0, false, false);
        acc1 = __builtin_amdgcn_wmma_f32_16x16x4_f32(false, a1, false, b11, (short)0, acc1, false, false);
        acc1 = __builtin_amdgcn_wmma_f32_16x16x4_f32(false, a2, false, b21, (short)0, acc1, false, false);
    }

    // Epilogue: C/D layout — VGPR r: lanes 0-15 hold (M=r, N=lane),
    //                                 lanes 16-31 hold (M=r+8, N=lane-16)
#pragma unroll
    for (int nt = 0; nt < 2; ++nt) {
        v8f acc = nt ? acc1 : acc0;
        int k = nt * 16 + Ml;
        if (k < NN) {
            float c = c0[i * NN + k];
#pragma unroll
            for (int r = 0; r < 8; ++r) {
                int b = b0 + r + hi * 8;
                if (b < B) {
                    long long e = ((long long)b * NN + i) * NN + k;
                    float dv = flow[e * TT + (TT - 1)];
                    out[e] = dv * c + acc[r];
                }
            }
        }
    }
}

// ---------------------------------------------------------------------------
extern "C" void kernel_launch(void* const* d_in, const int* in_sizes, int n_in,
                              void* d_out, int out_size, void* d_ws, size_t ws_size,
                              hipStream_t stream)
{
    const float* flow = (const float*)d_in[0];   // [B, N, N, T] f32
    const int*   adj  = (const int*)  d_in[1];   // [N, N] i32
    const float* W1   = (const float*)d_in[2];   // [N, N, N]
    const float* W2   = (const float*)d_in[3];   // [N, N, N]
    const float* ain  = (const float*)d_in[4];   // [N, N, 2]
    const float* aout = (const float*)d_in[5];   // [1, N, N, 2]
    float* outp = (float*)d_out;                 // [B, N, N, 1]

    const int NPAD = NN * KP * MP;               // padded matrix size
    float* ws  = (float*)d_ws;
    float* A1p = ws;
    float* A2p = ws + NPAD;
    float* c0  = ws + 2 * NPAD;

    int B = in_sizes[0] / (NN * NN * TT);        // 8192

    resgat_prep<<<(NN * KP + 127) / 128, 128, 0, stream>>>(
        adj, W1, W2, ain, aout, A1p, A2p, c0);

    int btiles      = (B + 15) / 16;
    int totalTiles  = btiles * NN;               // 11776 waves
    int wavesPerBlk = 8;                         // 256 threads
    int blocks = (totalTiles + wavesPerBlk - 1) / wavesPerBlk;

    resgat_main<<<blocks, wavesPerBlk * 32, 0, stream>>>(
        flow, A1p, A2p, c0, outp, B, totalTiles);
}